// GCN_33105607917776
// MI455X (gfx1250) — compile-verified
//
#include <hip/hip_runtime.h>
#include <math.h>

#define EPS 1e-5f

typedef float v2f __attribute__((ext_vector_type(2)));
typedef float v8f __attribute__((ext_vector_type(8)));

static constexpr int   cN = 100000;
static constexpr int   cE = 1600000;
static constexpr int   cG = 512;
static constexpr int   cFIN = 128;
static constexpr int   cD = 64;
static constexpr int   cL = 256;

#define GRID1(n, b) ((unsigned)(((long)(n) + (b) - 1) / (b)))

// ---------------------------------------------------------------------------
// Generic WMMA f32 GEMM: C[M x Nc] = A[M x K] @ B[K x Nc] (+ bias[col]).
// One wave computes one 16x16 tile via V_WMMA_F32_16X16X4_F32 over K/4 steps.
// Optional epilogue: accumulate per-column sum / sum-of-squares (for BN stats).
// M % 16 == 0, K % 4 == 0, Nc % 16 == 0 (true for all call sites).
// ---------------------------------------------------------------------------
__global__ void wmma_gemm_f32(const float* __restrict__ A, const float* __restrict__ B,
                              const float* __restrict__ bias, float* __restrict__ C,
                              float* __restrict__ stat_sum, float* __restrict__ stat_sq,
                              int M, int K, int Nc, int store)
{
    int wave = (int)((blockIdx.x * blockDim.x + threadIdx.x) >> 5);
    int lane = threadIdx.x & 31;
    int ntn  = Nc >> 4;
    int tm   = wave / ntn;
    int tn   = wave % ntn;
    if (tm * 16 >= M) return;
    int m0 = tm * 16, n0 = tn * 16;
    int half = lane >> 4;     // lanes 16..31 hold K+2,K+3 pair
    int l    = lane & 15;

    v8f acc = {};
    const float* Arow = A + (size_t)(m0 + l) * K;
    for (int k = 0; k < K; k += 4) {
        int ka = k + half * 2;
        v2f a, b;
        a.x = Arow[ka];
        a.y = Arow[ka + 1];
        b.x = B[(size_t)ka * Nc + n0 + l];
        b.y = B[(size_t)(ka + 1) * Nc + n0 + l];
        acc = __builtin_amdgcn_wmma_f32_16x16x4_f32(false, a, false, b,
                                                    (short)0, acc, false, false);
    }

    int col = n0 + l;
    float bv = bias ? bias[col] : 0.0f;
    float s = 0.0f, sq = 0.0f;
#pragma unroll
    for (int i = 0; i < 8; ++i) {
        int row = m0 + i + half * 8;   // C/D layout: VGPR i -> M=i (lo half), M=i+8 (hi half)
        float v = acc[i] + bv;
        if (store) C[(size_t)row * Nc + col] = v;
        s += v; sq += v * v;
    }
    if (stat_sum) {
        atomicAdd(&stat_sum[col], s);
        atomicAdd(&stat_sq[col],  sq);
    }
}

// ---------------------------------------------------------------------------
// Fills
// ---------------------------------------------------------------------------
__global__ void fill_f32(float* p, float v, long n) {
    long t = blockIdx.x * (long)blockDim.x + threadIdx.x;
    if (t < n) p[t] = v;
}
__global__ void fill_u32(unsigned* p, unsigned v, long n) {
    long t = blockIdx.x * (long)blockDim.x + threadIdx.x;
    if (t < n) p[t] = v;
}

// ---------------------------------------------------------------------------
// Degree / norm precompute
// ---------------------------------------------------------------------------
__global__ void deg_edge_kernel(const float* __restrict__ ea, const int* __restrict__ col,
                                float* __restrict__ deg, int E) {
    int e = blockIdx.x * blockDim.x + threadIdx.x;
    if (e < E) atomicAdd(&deg[col[e]], ea[e]);
}
__global__ void node_init_kernel(float* __restrict__ deg, float* __restrict__ cnt,
                                 const int* __restrict__ batch, int N) {
    int n = blockIdx.x * blockDim.x + threadIdx.x;
    if (n < N) {
        deg[n] += 1.0f;                       // self-loop weight 1 (prev kernel done)
        atomicAdd(&cnt[batch[n]], 1.0f);
    }
}
__global__ void dinv_kernel(float* deg, int N) {
    int n = blockIdx.x * blockDim.x + threadIdx.x;
    if (n < N) { float d = deg[n]; deg[n] = d > 0.0f ? rsqrtf(d) : 0.0f; }
}
__global__ void norm_kernel(const int* __restrict__ row, const int* __restrict__ col,
                            const float* __restrict__ ea, const float* __restrict__ dinv,
                            float* __restrict__ nrm, int E, int N) {
    int e = blockIdx.x * blockDim.x + threadIdx.x;
    if (e >= E + N) return;
    if (e < E) nrm[e] = dinv[row[e]] * ea[e] * dinv[col[e]];
    else       { int n = e - E; nrm[e] = dinv[n] * dinv[n]; }
}

// ---------------------------------------------------------------------------
// GCN aggregation: out[col] += norm[e] * xw[row]  (float4 per thread, 64 feats)
// ---------------------------------------------------------------------------
__global__ void scatter_kernel(const float* __restrict__ xw, const float* __restrict__ nrm,
                               const int* __restrict__ row, const int* __restrict__ col,
                               float* __restrict__ out, int E, int N) {
    long t = blockIdx.x * (long)blockDim.x + threadIdx.x;
    long total = (long)(E + N) * 16;
    if (t >= total) return;
    int  dg = (int)(t & 15);
    long e  = t >> 4;
    int r, c;
    if (e < E) { r = row[e]; c = col[e]; } else { r = c = (int)(e - E); }
    float w = nrm[e];
    float4 v = ((const float4*)(xw + (size_t)r * 64))[dg];
    float* o = out + (size_t)c * 64 + dg * 4;
    atomicAdd(o + 0, w * v.x);
    atomicAdd(o + 1, w * v.y);
    atomicAdd(o + 2, w * v.z);
    atomicAdd(o + 3, w * v.w);
}

__global__ void bias_add_kernel(float* __restrict__ h, const float* __restrict__ b,
                                long total, int Nc) {
    long t = blockIdx.x * (long)blockDim.x + threadIdx.x;
    if (t < total) h[t] += b[t % Nc];
}

// ---------------------------------------------------------------------------
// GraphNorm (per-graph mean/var over D=64 features) + ReLU
// ---------------------------------------------------------------------------
__global__ void gn_sum_kernel(const float* __restrict__ h, const int* __restrict__ batch,
                              float* __restrict__ gmean, long total) {
    long t = blockIdx.x * (long)blockDim.x + threadIdx.x;
    if (t >= total) return;
    int n = (int)(t >> 6), d = (int)(t & 63);
    atomicAdd(&gmean[(size_t)batch[n] * 64 + d], h[t]);
}
__global__ void div_count_kernel(float* __restrict__ buf, const float* __restrict__ cnt, int total) {
    int i = blockIdx.x * blockDim.x + threadIdx.x;
    if (i < total) buf[i] /= fmaxf(cnt[i >> 6], 1.0f);
}
__global__ void gn_var_kernel(const float* __restrict__ h, const int* __restrict__ batch,
                              const float* __restrict__ gmean, const float* __restrict__ alpha,
                              float* __restrict__ gvar, long total) {
    long t = blockIdx.x * (long)blockDim.x + threadIdx.x;
    if (t >= total) return;
    int n = (int)(t >> 6), d = (int)(t & 63), g = batch[n];
    float sub = h[t] - alpha[d] * gmean[(size_t)g * 64 + d];
    atomicAdd(&gvar[(size_t)g * 64 + d], sub * sub);
}
__global__ void gn_apply_kernel(float* __restrict__ h, const int* __restrict__ batch,
                                const float* __restrict__ gmean, const float* __restrict__ gvar,
                                const float* __restrict__ gamma, const float* __restrict__ beta,
                                const float* __restrict__ alpha, long total) {
    long t = blockIdx.x * (long)blockDim.x + threadIdx.x;
    if (t >= total) return;
    int n = (int)(t >> 6), d = (int)(t & 63), g = batch[n];
    float sub = h[t] - alpha[d] * gmean[(size_t)g * 64 + d];
    float v = gamma[d] * sub * rsqrtf(gvar[(size_t)g * 64 + d] + EPS) + beta[d];
    h[t] = fmaxf(v, 0.0f);
}

// ---------------------------------------------------------------------------
// BatchNorm helpers (column stats accumulated by GEMM epilogue)
// ---------------------------------------------------------------------------
__global__ void bn_finalize_kernel(float* s, float* sq, float rows, int Nc) {
    int c = blockIdx.x * blockDim.x + threadIdx.x;
    if (c < Nc) {
        float m = s[c] / rows;
        float v = sq[c] / rows - m * m;   // biased variance (jnp.var default)
        s[c] = m; sq[c] = v;
    }
}
__global__ void bn_apply_kernel(float* __restrict__ z, const float* __restrict__ mean,
                                const float* __restrict__ var, const float* __restrict__ g,
                                const float* __restrict__ b, long total, int Nc) {
    long t = blockIdx.x * (long)blockDim.x + threadIdx.x;
    if (t >= total) return;
    int c = (int)(t % Nc);
    float v = g[c] * (z[t] - mean[c]) * rsqrtf(var[c] + EPS) + b[c];
    z[t] = fmaxf(v, 0.0f);
}

// ---------------------------------------------------------------------------
// Fused gate: recompute g1 = h@W1+b1 (wave per node), BN+ReLU, dot with w2
// ---------------------------------------------------------------------------
__global__ void gate_kernel(const float* __restrict__ h, const float* __restrict__ W1,
                            const float* __restrict__ b1, const float* __restrict__ bn_g,
                            const float* __restrict__ bn_b, const float* __restrict__ cmean,
                            const float* __restrict__ cvar, const float* __restrict__ w2,
                            const float* __restrict__ b2, float* __restrict__ gate, int N) {
    int wave = (int)((blockIdx.x * blockDim.x + threadIdx.x) >> 5);
    int lane = threadIdx.x & 31;
    if (wave >= N) return;
    const float* hr = h + (size_t)wave * 64;
    float acc0 = b1[lane], acc1 = b1[lane + 32], acc2 = b1[lane + 64], acc3 = b1[lane + 96];
    for (int k = 0; k < 64; ++k) {
        float hv = hr[k];
        const float* wr = W1 + (size_t)k * 128;
        acc0 += hv * wr[lane];
        acc1 += hv * wr[lane + 32];
        acc2 += hv * wr[lane + 64];
        acc3 += hv * wr[lane + 96];
    }
    float part = 0.0f;
    float accs[4] = {acc0, acc1, acc2, acc3};
#pragma unroll
    for (int j = 0; j < 4; ++j) {
        int c = lane + j * 32;
        float v = bn_g[c] * (accs[j] - cmean[c]) * rsqrtf(cvar[c] + EPS) + bn_b[c];
        v = fmaxf(v, 0.0f);
        part += v * w2[c];
    }
    for (int off = 16; off; off >>= 1) part += __shfl_xor(part, off, 32);
    if (lane == 0) gate[wave] = part + b2[0];
}

// ---------------------------------------------------------------------------
// Per-graph softmax + attention pooling
// ---------------------------------------------------------------------------
__global__ void seg_max_kernel(const float* __restrict__ gate, const int* __restrict__ batch,
                               unsigned* __restrict__ gmax, int N) {
    int n = blockIdx.x * blockDim.x + threadIdx.x;
    if (n >= N) return;
    float v = gate[n];
    int g = batch[n];
    if (v >= 0.0f) atomicMax((int*)&gmax[g], __float_as_int(v));
    else           atomicMin(&gmax[g], __float_as_uint(v));
}
__global__ void exp_sum_kernel(float* __restrict__ gate, const int* __restrict__ batch,
                               const unsigned* __restrict__ gmax, float* __restrict__ denom, int N) {
    int n = blockIdx.x * blockDim.x + threadIdx.x;
    if (n >= N) return;
    int g = batch[n];
    float e = expf(gate[n] - __uint_as_float(gmax[g]));
    gate[n] = e;
    atomicAdd(&denom[g], e);
}
__global__ void pool_kernel(const float* __restrict__ e, const float* __restrict__ denom,
                            const float* __restrict__ h, const int* __restrict__ batch,
                            float* __restrict__ pooled, long total) {
    long t = blockIdx.x * (long)blockDim.x + threadIdx.x;
    if (t >= total) return;
    int n = (int)(t >> 6), d = (int)(t & 63), g = batch[n];
    float a = e[n] / denom[g];
    atomicAdd(&pooled[(size_t)g * 64 + d], a * h[t]);
}

// ---------------------------------------------------------------------------
// Output head: [G,256] @ [256,2] + bias
// ---------------------------------------------------------------------------
__global__ void out_kernel(const float* __restrict__ z, const float* __restrict__ W,
                           const float* __restrict__ b, float* __restrict__ out, int G) {
    int t = blockIdx.x * blockDim.x + threadIdx.x;
    if (t >= G * 2) return;
    int g = t >> 1, o = t & 1;
    const float* zr = z + (size_t)g * 256;
    float s = b[o];
    for (int k = 0; k < 256; ++k) s += zr[k] * W[k * 2 + o];
    out[t] = s;
}

// ---------------------------------------------------------------------------
static inline void launch_gemm(const float* A, const float* B, const float* bias,
                               float* C, float* ssum, float* ssq,
                               int M, int K, int Nc, int store, hipStream_t s) {
    long threads = (long)(M / 16) * (Nc / 16) * 32;
    wmma_gemm_f32<<<GRID1(threads, 256), 256, 0, s>>>(A, B, bias, C, ssum, ssq, M, K, Nc, store);
}

extern "C" void kernel_launch(void* const* d_in, const int* in_sizes, int n_in,
                              void* d_out, int out_size, void* d_ws, size_t ws_size,
                              hipStream_t stream) {
    const int N = cN, E = cE, G = cG;
    const float* x        = (const float*)d_in[0];
    const float* ea       = (const float*)d_in[1];
    const int*   ei       = (const int*)d_in[2];
    const int*   batch    = (const int*)d_in[3];
    const float* conv1_w  = (const float*)d_in[4];
    const float* conv1_b  = (const float*)d_in[5];
    const float* gn1_g    = (const float*)d_in[6];
    const float* gn1_b    = (const float*)d_in[7];
    const float* gn1_a    = (const float*)d_in[8];
    const float* conv2_w  = (const float*)d_in[9];
    const float* conv2_b  = (const float*)d_in[10];
    const float* gn2_g    = (const float*)d_in[11];
    const float* gn2_b    = (const float*)d_in[12];
    const float* gn2_a    = (const float*)d_in[13];
    const float* gate_w1  = (const float*)d_in[14];
    const float* gate_b1  = (const float*)d_in[15];
    const float* gate_bng = (const float*)d_in[16];
    const float* gate_bnb = (const float*)d_in[17];
    const float* gate_w2  = (const float*)d_in[18];
    const float* gate_b2  = (const float*)d_in[19];
    const float* lin0_w   = (const float*)d_in[20];
    const float* lin0_b   = (const float*)d_in[21];
    const float* bn0_g    = (const float*)d_in[22];
    const float* bn0_b    = (const float*)d_in[23];
    const float* lin1_w   = (const float*)d_in[24];
    const float* lin1_b   = (const float*)d_in[25];
    const float* bn1_g    = (const float*)d_in[26];
    const float* bn1_b    = (const float*)d_in[27];
    const float* lin2_w   = (const float*)d_in[28];
    const float* lin2_b   = (const float*)d_in[29];
    const float* bn2_g    = (const float*)d_in[30];
    const float* bn2_b    = (const float*)d_in[31];
    const float* linout_w = (const float*)d_in[32];
    const float* linout_b = (const float*)d_in[33];

    const int* row = ei;        // edge_index[0]
    const int* col = ei + E;    // edge_index[1]

    // workspace carve-up (floats)
    float* ws = (float*)d_ws;
    size_t o = 0;
    float* bufA   = ws + o; o += (size_t)N * 64;   // xw / h@W2
    float* bufB   = ws + o; o += (size_t)N * 64;   // conv out / h
    float* dinv   = ws + o; o += N;                // deg -> dinv (in place)
    float* nrm    = ws + o; o += (size_t)E + N;
    float* cnt    = ws + o; o += G;
    float* gmean  = ws + o; o += (size_t)G * 64;
    float* gvar   = ws + o; o += (size_t)G * 64;
    float* colsum = ws + o; o += 256;
    float* colsq  = ws + o; o += 256;
    float* gate   = ws + o; o += N;
    unsigned* gmax = (unsigned*)(ws + o); o += G;
    float* denom  = ws + o; o += G;
    float* pooled = ws + o; o += (size_t)G * 64;
    float* z0     = ws + o; o += (size_t)G * 256;
    float* z1     = ws + o; o += (size_t)G * 256;

    const long ND = (long)N * 64;

    // --- degree / dinv / norm -------------------------------------------------
    fill_f32<<<GRID1(N, 256), 256, 0, stream>>>(dinv, 0.0f, N);
    fill_f32<<<GRID1(G, 256), 256, 0, stream>>>(cnt, 0.0f, G);
    deg_edge_kernel<<<GRID1(E, 256), 256, 0, stream>>>(ea, col, dinv, E);
    node_init_kernel<<<GRID1(N, 256), 256, 0, stream>>>(dinv, cnt, batch, N);
    dinv_kernel<<<GRID1(N, 256), 256, 0, stream>>>(dinv, N);
    norm_kernel<<<GRID1(E + N, 256), 256, 0, stream>>>(row, col, ea, dinv, nrm, E, N);

    // --- conv1: bufA = x @ W1 ; bufB = scatter + bias ; GraphNorm1 + ReLU -----
    launch_gemm(x, conv1_w, nullptr, bufA, nullptr, nullptr, N, 128, 64, 1, stream);
    fill_f32<<<GRID1(ND, 256), 256, 0, stream>>>(bufB, 0.0f, ND);
    scatter_kernel<<<GRID1((long)(E + N) * 16, 256), 256, 0, stream>>>(bufA, nrm, row, col, bufB, E, N);
    bias_add_kernel<<<GRID1(ND, 256), 256, 0, stream>>>(bufB, conv1_b, ND, 64);
    fill_f32<<<GRID1(G * 64, 256), 256, 0, stream>>>(gmean, 0.0f, (long)G * 64);
    gn_sum_kernel<<<GRID1(ND, 256), 256, 0, stream>>>(bufB, batch, gmean, ND);
    div_count_kernel<<<GRID1(G * 64, 256), 256, 0, stream>>>(gmean, cnt, G * 64);
    fill_f32<<<GRID1(G * 64, 256), 256, 0, stream>>>(gvar, 0.0f, (long)G * 64);
    gn_var_kernel<<<GRID1(ND, 256), 256, 0, stream>>>(bufB, batch, gmean, gn1_a, gvar, ND);
    div_count_kernel<<<GRID1(G * 64, 256), 256, 0, stream>>>(gvar, cnt, G * 64);
    gn_apply_kernel<<<GRID1(ND, 256), 256, 0, stream>>>(bufB, batch, gmean, gvar, gn1_g, gn1_b, gn1_a, ND);

    // --- conv2 ----------------------------------------------------------------
    launch_gemm(bufB, conv2_w, nullptr, bufA, nullptr, nullptr, N, 64, 64, 1, stream);
    fill_f32<<<GRID1(ND, 256), 256, 0, stream>>>(bufB, 0.0f, ND);
    scatter_kernel<<<GRID1((long)(E + N) * 16, 256), 256, 0, stream>>>(bufA, nrm, row, col, bufB, E, N);
    bias_add_kernel<<<GRID1(ND, 256), 256, 0, stream>>>(bufB, conv2_b, ND, 64);
    fill_f32<<<GRID1(G * 64, 256), 256, 0, stream>>>(gmean, 0.0f, (long)G * 64);
    gn_sum_kernel<<<GRID1(ND, 256), 256, 0, stream>>>(bufB, batch, gmean, ND);
    div_count_kernel<<<GRID1(G * 64, 256), 256, 0, stream>>>(gmean, cnt, G * 64);
    fill_f32<<<GRID1(G * 64, 256), 256, 0, stream>>>(gvar, 0.0f, (long)G * 64);
    gn_var_kernel<<<GRID1(ND, 256), 256, 0, stream>>>(bufB, batch, gmean, gn2_a, gvar, ND);
    div_count_kernel<<<GRID1(G * 64, 256), 256, 0, stream>>>(gvar, cnt, G * 64);
    gn_apply_kernel<<<GRID1(ND, 256), 256, 0, stream>>>(bufB, batch, gmean, gvar, gn2_g, gn2_b, gn2_a, ND);
    // bufB now holds h2 [N,64]

    // --- gate: BN stats via WMMA GEMM epilogue (g1 never materialized) --------
    fill_f32<<<1, 256, 0, stream>>>(colsum, 0.0f, 128);
    fill_f32<<<1, 256, 0, stream>>>(colsq, 0.0f, 128);
    launch_gemm(bufB, gate_w1, gate_b1, nullptr, colsum, colsq, N, 64, 128, 0, stream);
    bn_finalize_kernel<<<1, 256, 0, stream>>>(colsum, colsq, (float)N, 128);
    gate_kernel<<<GRID1((long)N * 32, 256), 256, 0, stream>>>(bufB, gate_w1, gate_b1, gate_bng,
                                                              gate_bnb, colsum, colsq, gate_w2,
                                                              gate_b2, gate, N);
    fill_u32<<<GRID1(G, 256), 256, 0, stream>>>(gmax, 0xFF800000u, G);   // -inf bits
    fill_f32<<<GRID1(G, 256), 256, 0, stream>>>(denom, 0.0f, G);
    seg_max_kernel<<<GRID1(N, 256), 256, 0, stream>>>(gate, batch, gmax, N);
    exp_sum_kernel<<<GRID1(N, 256), 256, 0, stream>>>(gate, batch, gmax, denom, N);
    fill_f32<<<GRID1(G * 64, 256), 256, 0, stream>>>(pooled, 0.0f, (long)G * 64);
    pool_kernel<<<GRID1(ND, 256), 256, 0, stream>>>(gate, denom, bufB, batch, pooled, ND);

    // --- MLP head -------------------------------------------------------------
    const long GL = (long)G * 256;
    fill_f32<<<1, 256, 0, stream>>>(colsum, 0.0f, 256);
    fill_f32<<<1, 256, 0, stream>>>(colsq, 0.0f, 256);
    launch_gemm(pooled, lin0_w, lin0_b, z0, colsum, colsq, G, 64, 256, 1, stream);
    bn_finalize_kernel<<<1, 256, 0, stream>>>(colsum, colsq, (float)G, 256);
    bn_apply_kernel<<<GRID1(GL, 256), 256, 0, stream>>>(z0, colsum, colsq, bn0_g, bn0_b, GL, 256);

    fill_f32<<<1, 256, 0, stream>>>(colsum, 0.0f, 256);
    fill_f32<<<1, 256, 0, stream>>>(colsq, 0.0f, 256);
    launch_gemm(z0, lin1_w, lin1_b, z1, colsum, colsq, G, 256, 256, 1, stream);
    bn_finalize_kernel<<<1, 256, 0, stream>>>(colsum, colsq, (float)G, 256);
    bn_apply_kernel<<<GRID1(GL, 256), 256, 0, stream>>>(z1, colsum, colsq, bn1_g, bn1_b, GL, 256);

    fill_f32<<<1, 256, 0, stream>>>(colsum, 0.0f, 256);
    fill_f32<<<1, 256, 0, stream>>>(colsq, 0.0f, 256);
    launch_gemm(z1, lin2_w, lin2_b, z0, colsum, colsq, G, 256, 256, 1, stream);
    bn_finalize_kernel<<<1, 256, 0, stream>>>(colsum, colsq, (float)G, 256);
    bn_apply_kernel<<<GRID1(GL, 256), 256, 0, stream>>>(z0, colsum, colsq, bn2_g, bn2_b, GL, 256);

    out_kernel<<<GRID1(G * 2, 256), 256, 0, stream>>>(z0, linout_w, linout_b, (float*)d_out, G);
}